// GraphSAGE_37752762532479
// MI455X (gfx1250) — compile-verified
//
#include <hip/hip_runtime.h>
#include <hip/hip_bf16.h>
#include <stdint.h>

typedef __attribute__((ext_vector_type(2))) float v2f;
typedef __attribute__((ext_vector_type(8))) float v8f;

#define N_NODES 50000
#define N_EDGES 800000
#define IN_DIM  128
#define HID_DIM 256
#define OUT_DIM 128
#define LDS_PAD 4          // 4-float row pad -> bank = (4*lane)%64, conflict-free

// ---------------------------------------------------------------------------
// Zero a float region (grid-stride).
// ---------------------------------------------------------------------------
__global__ void zero_f32(float* __restrict__ p, size_t n) {
    size_t i = (size_t)blockIdx.x * blockDim.x + threadIdx.x;
    size_t stride = (size_t)gridDim.x * blockDim.x;
    for (; i < n; i += stride) p[i] = 0.0f;
}

// ---------------------------------------------------------------------------
// Edge scatter: one wave32 per edge. Lane l handles channels [4l, 4l+4) (and
// +128 for D=256): float4 gather from feat[src] (contiguous 512B per wave),
// 4x global_atomic_add_f32 into msg[dst] (contiguous span per wave). The msg
// buffers (25.6/51.2 MB) are L2-resident, so the atomic RMW stays on-chip.
// ---------------------------------------------------------------------------
__global__ void scatter_sum(const long long* __restrict__ src,
                            const long long* __restrict__ dst,
                            const float* __restrict__ feat,
                            float* __restrict__ msg,
                            float* __restrict__ deg,
                            int E, int D, int addDeg)
{
    int wave = (int)(((size_t)blockIdx.x * blockDim.x + threadIdx.x) >> 5);
    int lane = threadIdx.x & 31;
    if (wave >= E) return;
    long long s = src[wave];
    long long d = dst[wave];
    const float* fr = feat + (size_t)s * D;
    float* mr = msg + (size_t)d * D;
    for (int c = lane * 4; c < D; c += 128) {
        const float4 v = *(const float4*)(fr + c);
        atomicAdd(mr + c + 0, v.x);
        atomicAdd(mr + c + 1, v.y);
        atomicAdd(mr + c + 2, v.z);
        atomicAdd(mr + c + 3, v.w);
    }
    if (addDeg && lane == 0) atomicAdd(deg + d, 1.0f);
}

// ---------------------------------------------------------------------------
// deg -> 1/max(deg, 1)  (in place)
// ---------------------------------------------------------------------------
__global__ void invdeg_k(float* __restrict__ deg, int n) {
    int i = blockIdx.x * blockDim.x + threadIdx.x;
    if (i < n) deg[i] = 1.0f / fmaxf(deg[i], 1.0f);
}

// ---------------------------------------------------------------------------
// Fused dual-GEMM via V_WMMA_F32_16X16X4_F32 (wave32 fp32 matrix path):
//   out[M,Nw] = (msg*invdeg) @ Wl^T + xin @ Wr^T + bias   (+ optional ReLU)
//
// Block = 256 threads = 8 waves sharing one N-tile; the block stages the two
// 16xK weight slabs into LDS once (coalesced float4 reads, padded rows ->
// conflict-free ds_load fragment reads). Each wave computes one 16x16 output
// tile. The K-loop runs in chunks of 32 with explicit register prefetch:
// all 16 A fragments (global b64) + 16 B fragments (LDS) of a chunk are
// issued before the 16 chained WMMAs, keeping many loads in flight instead
// of a wait-0 stall per WMMA pair.
//
// ISA VGPR layouts (CDNA5 ISA):
//   A 16x4 : lanes 0-15 -> M=lane, VGPR{0,1}=K{0,1}; lanes 16-31 -> K{2,3}
//   B 4x16 : lanes 0-15 -> N=lane, VGPR{0,1}=K{0,1}; lanes 16-31 -> K{2,3}
//   C/D    : VGPR i -> M = i + 8*(lane>=16), N = lane&15
// ---------------------------------------------------------------------------
template<int K>
__global__ void sage_gemm_wmma(const float* __restrict__ msg,
                               const float* __restrict__ xin,
                               const float* __restrict__ invdeg,
                               const float* __restrict__ Wl,  // [Nw, K] row-major
                               const float* __restrict__ Wr,  // [Nw, K] row-major
                               const float* __restrict__ bias,
                               float* __restrict__ outp,
                               int M, int Nw, int relu)
{
    constexpr int ldsStride = K + LDS_PAD;
    __shared__ float sWl[16 * ldsStride];
    __shared__ float sWr[16 * ldsStride];

    const int ntn   = Nw >> 4;
    const int tn    = blockIdx.x % ntn;
    const int chunk = blockIdx.x / ntn;     // 8 M-tiles per block

    // ---- cooperative stage of both weight slabs for column tile tn ----
    const float* gWl = Wl + (size_t)(tn * 16) * K;
    const float* gWr = Wr + (size_t)(tn * 16) * K;
    #pragma unroll
    for (int idx = 0; idx < 16 * K; idx += 256 * 4) {
        int t = idx + threadIdx.x * 4;
        int row = t / K;
        int col = t - row * K;              // multiple of 4 (K % 4 == 0)
        *(float4*)(sWl + row * ldsStride + col) =
            *(const float4*)(gWl + (size_t)row * K + col);
        *(float4*)(sWr + row * ldsStride + col) =
            *(const float4*)(gWr + (size_t)row * K + col);
    }
    __syncthreads();

    const int wave = threadIdx.x >> 5;
    const int lane = threadIdx.x & 31;
    const int tm   = chunk * 8 + wave;
    if (tm >= (M >> 4)) return;             // wave-uniform: EXEC stays all-1s

    const int half = lane >> 4;             // 0 or 1 -> K offset {0, 2}
    const int l15  = lane & 15;

    const int rowA = tm * 16 + l15;         // A-fragment row for this lane
    const float idg = invdeg[rowA];
    const float* aR1 = msg + (size_t)rowA * K;   // lin_l operand (mean agg)
    const float* aR2 = xin + (size_t)rowA * K;   // lin_r operand (self)
    const float* sB1 = sWl + l15 * ldsStride;    // B fragment rows (N = l15)
    const float* sB2 = sWr + l15 * ldsStride;

    v8f acc = {};
    #pragma unroll 1
    for (int k0 = 0; k0 < K; k0 += 32) {    // 8 k-steps per chunk
        v2f a1[8], a2[8], bf1[8], bf2[8];
        #pragma unroll
        for (int j = 0; j < 8; ++j) {       // issue 16 global b64 loads
            int ka = k0 + j * 4 + half * 2;
            a1[j].x = aR1[ka]; a1[j].y = aR1[ka + 1];
            a2[j].x = aR2[ka]; a2[j].y = aR2[ka + 1];
        }
        #pragma unroll
        for (int j = 0; j < 8; ++j) {       // issue 16 LDS loads
            int ka = k0 + j * 4 + half * 2;
            bf1[j].x = sB1[ka]; bf1[j].y = sB1[ka + 1];
            bf2[j].x = sB2[ka]; bf2[j].y = sB2[ka + 1];
        }
        #pragma unroll
        for (int j = 0; j < 8; ++j) {       // drain: 16 chained WMMAs
            v2f as; as.x = a1[j].x * idg; as.y = a1[j].y * idg;
            acc = __builtin_amdgcn_wmma_f32_16x16x4_f32(
                      false, as, false, bf1[j], (short)0, acc, false, false);
            acc = __builtin_amdgcn_wmma_f32_16x16x4_f32(
                      false, a2[j], false, bf2[j], (short)0, acc, false, false);
        }
    }

    const int colD = tn * 16 + l15;
    const int rowBase = tm * 16 + half * 8;
    const float bv = bias[colD];
    #pragma unroll
    for (int i = 0; i < 8; ++i) {
        float v = acc[i] + bv;
        if (relu) v = fmaxf(v, 0.0f);
        outp[(size_t)(rowBase + i) * Nw + colD] = v;
    }
}

// ---------------------------------------------------------------------------
// Launch: zero -> scatter(x,+deg) -> invdeg -> GEMM1(ReLU) -> zero ->
//         scatter(h) -> GEMM2 -> d_out.   All on `stream`, graph-capturable.
// ---------------------------------------------------------------------------
extern "C" void kernel_launch(void* const* d_in, const int* in_sizes, int n_in,
                              void* d_out, int out_size, void* d_ws, size_t ws_size,
                              hipStream_t stream) {
    const float*      x    = (const float*)d_in[0];
    const long long*  ei   = (const long long*)d_in[1];   // int64 [2, E]
    const float*      W_l1 = (const float*)d_in[2];       // [256,128]
    const float*      W_r1 = (const float*)d_in[3];       // [256,128]
    const float*      b1   = (const float*)d_in[4];       // [256]
    const float*      W_l2 = (const float*)d_in[5];       // [128,256]
    const float*      W_r2 = (const float*)d_in[6];       // [128,256]
    const float*      b2   = (const float*)d_in[7];       // [128]
    float*            out  = (float*)d_out;               // [50000,128]

    const long long* src = ei;            // row 0
    const long long* dst = ei + N_EDGES;  // row 1

    // Workspace layout (floats): deg(50016 pad) | buf1(50000*256) | buf2(50000*256)
    float* deg  = (float*)d_ws;
    float* buf1 = deg + 50016;                      // msg1 / msg2 (reused)
    float* buf2 = buf1 + (size_t)N_NODES * HID_DIM; // h (layer-1 output)

    const int ZB = 1024;
    const int scatterBlocks = (N_EDGES * 32) / 256;           // 100000
    const int mChunks = (N_NODES / 16 + 7) / 8;               // 391

    // ---- Layer 1 ----
    zero_f32<<<ZB, 256, 0, stream>>>(deg, (size_t)50016 + (size_t)N_NODES * IN_DIM);
    scatter_sum<<<scatterBlocks, 256, 0, stream>>>(src, dst, x, buf1, deg,
                                                   N_EDGES, IN_DIM, 1);
    invdeg_k<<<(N_NODES + 255) / 256, 256, 0, stream>>>(deg, N_NODES);

    {   // h = relu((msg1/deg) @ W_l1^T + x @ W_r1^T + b1)   [50000, 256]
        int ntn = HID_DIM / 16;                                // 16
        sage_gemm_wmma<IN_DIM><<<ntn * mChunks, 256, 0, stream>>>(
            buf1, x, deg, W_l1, W_r1, b1, buf2,
            N_NODES, HID_DIM, /*relu=*/1);
    }

    // ---- Layer 2 ----
    zero_f32<<<ZB, 256, 0, stream>>>(buf1, (size_t)N_NODES * HID_DIM);
    scatter_sum<<<scatterBlocks, 256, 0, stream>>>(src, dst, buf2, buf1, deg,
                                                   N_EDGES, HID_DIM, 0);
    {   // out = (msg2/deg) @ W_l2^T + h @ W_r2^T + b2       [50000, 128]
        int ntn = OUT_DIM / 16;                                // 8
        sage_gemm_wmma<HID_DIM><<<ntn * mChunks, 256, 0, stream>>>(
            buf1, buf2, deg, W_l2, W_r2, b2, out,
            N_NODES, HID_DIM, /*relu=*/0);
    }
}